// CrossEntropyLoss_mean_81518479278686
// MI455X (gfx1250) — compile-verified
//
#include <hip/hip_runtime.h>
#include <hip/hip_bf16.h>
#include <stdint.h>

// ---------------------------------------------------------------------------
// CrossEntropy-style loss, MI455X (gfx1250).
// Phase 1: rowwise logsumexp over V=32000 streamed via TDM (tensor_load_to_lds)
//          into double-buffered LDS; online (max,sum) accumulation; writes
//          lp_t[b,t] = scores[b,t,tgt] - logsumexp to workspace.
// Phase 2: single block: leaky-integrator scan, ragged softmax, weighted dot;
//          final 64-way reduction via V_WMMA_F32_16X16X4_F32 with a ones B.
// ---------------------------------------------------------------------------

typedef unsigned int u32x4 __attribute__((ext_vector_type(4)));
typedef int          i32x8 __attribute__((ext_vector_type(8)));
typedef int          i32x4 __attribute__((ext_vector_type(4)));
typedef float        v2f   __attribute__((ext_vector_type(2)));
typedef float        v8f   __attribute__((ext_vector_type(8)));

#define NT    256
#define CHUNK 4096            // floats per TDM chunk (16 KB)
#define VDIM  32000
#define TDIM  1024
#define BDIM  8

// Issue a 1-D TDM load of `elems` fp32 from global `gsrc` to LDS byte address
// `lds_addr`. D# layout per CDNA5 ISA ch.8:
//   group0: [1:0]=count=1, [63:32]=lds_addr, [120:64]=global_addr, [127:126]=type=2
//   group1: [17:16]=data_size=2 (4B), tensor_dim0=elems, tensor_dim1=1,
//           tile_dim0=elems, tile_dim1=0 (unused -> 1-D), strides benign.
__device__ __forceinline__ void tdm_load_1d(const float* gsrc,
                                            uint32_t lds_addr, int elems) {
  uint64_t ga = (uint64_t)(uintptr_t)gsrc;
  uint32_t e  = (uint32_t)elems;
  u32x4 g0;
  g0[0] = 1u;                                               // count=1, user desc
  g0[1] = lds_addr;                                         // lds_addr
  g0[2] = (uint32_t)ga;                                     // global_addr[31:0]
  g0[3] = ((uint32_t)(ga >> 32) & 0x01FFFFFFu) | 0x80000000u; // addr[56:32] | type=2
  i32x8 g1;
  g1[0] = 0x00020000;                                       // data_size=4B, no mask
  g1[1] = (int)((e & 0xFFFFu) << 16);                       // tensor_dim0[15:0]
  g1[2] = (int)(((e >> 16) & 0xFFFFu) | (1u << 16));        // tensor_dim0[31:16], tensor_dim1=1
  g1[3] = (int)((e & 0xFFFFu) << 16);                       // tile_dim0 = elems
  g1[4] = 0;                                                // tile_dim1=0, tile_dim2=0
  g1[5] = (int)e;                                           // tensor_dim0_stride lo
  g1[6] = 0;
  g1[7] = 0;
  i32x4 z4 = {0, 0, 0, 0};                                  // groups 2/3 unused (2-D max)
  i32x8 z8 = {0, 0, 0, 0, 0, 0, 0, 0};
  __builtin_amdgcn_tensor_load_to_lds(g0, g1, z4, z4, z8, 0);
}

__global__ void __launch_bounds__(NT)
phase1_logsumexp(const float* __restrict__ scores,
                 const int* __restrict__ target,
                 float* __restrict__ lp_out) {
  __shared__ float buf[2][CHUNK];
  __shared__ float wm[NT / 32];
  __shared__ float wsum[NT / 32];

  const int row = blockIdx.x;                 // 0 .. B*T-1
  const float* rowp = scores + (size_t)row * VDIM;
  const int tid = threadIdx.x;
  const int nch = (VDIM + CHUNK - 1) / CHUNK; // 8

  // Prologue: wave 0 issues chunk 0 into buf[0].
  if (tid < 32) {
    tdm_load_1d(rowp, (uint32_t)(uintptr_t)&buf[0][0], min(CHUNK, VDIM));
  }

  float m = -INFINITY;
  float s = 0.0f;
  for (int c = 0; c < nch; ++c) {
    const int cur = c & 1;
    if (tid < 32) __builtin_amdgcn_s_wait_tensorcnt(0);  // chunk c landed in LDS
    __syncthreads();                                     // publish to all waves
    if (tid < 32 && (c + 1) < nch) {                     // prefetch next chunk
      const int e = min(CHUNK, VDIM - (c + 1) * CHUNK);
      tdm_load_1d(rowp + (size_t)(c + 1) * CHUNK,
                  (uint32_t)(uintptr_t)&buf[1 - cur][0], e);
    }
    const int elems = min(CHUNK, VDIM - c * CHUNK);
    for (int i = tid; i < elems; i += NT) {
      float x = buf[cur][i];
      if (x > m) {                                       // online logsumexp
        s = s * __expf(m - x) + 1.0f;
        m = x;
      } else {
        s += __expf(x - m);
      }
    }
    __syncthreads();  // everyone done reading buf[cur] before it is refilled
  }

  // Intra-wave combine of (m, s).
  const int lane = tid & 31, wid = tid >> 5;
  for (int off = 16; off > 0; off >>= 1) {
    float om = __shfl_xor(m, off, 32);
    float os = __shfl_xor(s, off, 32);
    float M  = fmaxf(m, om);
    s = s * __expf(m - M) + os * __expf(om - M);
    m = M;
  }
  if (lane == 0) { wm[wid] = m; wsum[wid] = s; }
  __syncthreads();

  if (tid == 0) {
    float M = wm[0], S = wsum[0];
    for (int w = 1; w < NT / 32; ++w) {
      float m2 = wm[w], s2 = wsum[w];
      float Mn = fmaxf(M, m2);
      S = S * __expf(M - Mn) + s2 * __expf(m2 - Mn);
      M = Mn;
    }
    const int tgt = target[row];
    const float x = rowp[tgt];
    lp_out[row] = x - M - __logf(S);   // log_softmax gathered at target
  }
}

__global__ void __launch_bounds__(NT)
phase2_loss(const float* __restrict__ lp_in,
            const int* __restrict__ lengths,
            float* __restrict__ out) {
  __shared__ float props[BDIM * TDIM];   // 32 KB
  __shared__ int   len_s[BDIM];
  __shared__ float wred[64];             // padded partials for WMMA reduce

  const int tid = threadIdx.x;
  if (tid < BDIM) len_s[tid] = lengths[tid];
  if (tid < 64)   wred[tid]  = 0.0f;
  __syncthreads();

  // Leaky-integrator scan: props[b,0]=0.5; props[b,t]=0.3*prev + 0.7*p[b,t-1],
  // p = exp(masked lp) (padded positions contribute exp(0)=1, per reference).
  if (tid < BDIM) {
    const int b = tid;
    const int L = len_s[b];
    const float* lpr = lp_in + b * TDIM;
    float prev = 0.5f;
    props[b * TDIM] = prev;
    for (int t = 1; t < TDIM; ++t) {
      float lp = (t - 1 < L) ? lpr[t - 1] : 0.0f;
      prev = 0.3f * prev + 0.7f * __expf(lp);
      props[b * TDIM + t] = prev;
    }
  }
  __syncthreads();

  // Wave w handles row w: ragged softmax over valid prefix, dot with lp.
  {
    const int wid = tid >> 5, lane = tid & 31;
    const int b = wid;
    const int L = len_s[b];
    const float* pr  = props + b * TDIM;
    const float* lpr = lp_in + b * TDIM;
    float mx = -INFINITY;
    for (int t = lane; t < L; t += 32) mx = fmaxf(mx, pr[t]);
    for (int off = 16; off > 0; off >>= 1) mx = fmaxf(mx, __shfl_xor(mx, off, 32));
    float se = 0.0f, acc = 0.0f;
    for (int t = lane; t < L; t += 32) {
      float e = __expf(pr[t] - mx);
      se  += e;
      acc += lpr[t] * e;
    }
    for (int off = 16; off > 0; off >>= 1) {
      se  += __shfl_xor(se, off, 32);
      acc += __shfl_xor(acc, off, 32);
    }
    // sum_t lp*soft = (sum lp*exp) * L / (sum exp)
    if (lane == 0) wred[b] = acc * (float)L / se;
  }
  __syncthreads();

  // Final 64 -> 1 reduction via D = A x ones (V_WMMA_F32_16X16X4_F32).
  // Every A entry contributes once to each column of D summed over rows, so
  // per-lane sum of the 8 D regs + xor-16 add yields the grand total.
  if (tid < 32) {
    v2f a;  a[0] = wred[tid]; a[1] = wred[tid + 32];
    v2f bo; bo[0] = 1.0f;     bo[1] = 1.0f;
    v8f c = {};
    v8f d = __builtin_amdgcn_wmma_f32_16x16x4_f32(
        /*neg_a=*/false, a, /*neg_b=*/false, bo,
        /*c_mod=*/(short)0, c, /*reuse_a=*/false, /*reuse_b=*/false);
    float tsum = d[0] + d[1] + d[2] + d[3] + d[4] + d[5] + d[6] + d[7];
    tsum += __shfl_xor(tsum, 16, 32);
    if (tid == 0) {
      float tot = 0.0f;
      for (int b = 0; b < BDIM; ++b) tot += (float)len_s[b];
      out[0] = -tsum / tot;
    }
  }
}

extern "C" void kernel_launch(void* const* d_in, const int* in_sizes, int n_in,
                              void* d_out, int out_size, void* d_ws, size_t ws_size,
                              hipStream_t stream) {
  const float* scores  = (const float*)d_in[0];  // [B,T,V] fp32
  const int*   target  = (const int*)d_in[1];    // [B,T] int32
  const int*   lengths = (const int*)d_in[2];    // [B] int32
  float* lp_ws = (float*)d_ws;                   // B*T floats = 32 KB
  float* outp  = (float*)d_out;                  // 1 float

  phase1_logsumexp<<<BDIM * TDIM, NT, 0, stream>>>(scores, target, lp_ws);
  phase2_loss<<<1, NT, 0, stream>>>(lp_ws, lengths, outp);
}